// BandSelection_83442624627152
// MI455X (gfx1250) — compile-verified
//
#include <hip/hip_runtime.h>

#define BB    8
#define BAND  25
#define EMB   32
#define HEADS 8
#define HDIM  4
#define HT    128
#define HS    256
#define NT    (HT*HT)   /* 16384 */
#define NS    (HS*HS)   /* 65536 */
#define TPW   4         /* N-tiles per wave in the WMMA kernel */

typedef __attribute__((ext_vector_type(2))) float v2f;
typedef __attribute__((ext_vector_type(8))) float v8f;

/* ---- flat f32 output offsets (reference tuple flattening order) ---- */
#define OFF_TARR  0
#define SZ_TARR   (4*BB*3*NT)              /* 1,572,864 */
#define OFF_SARR  (OFF_TARR + SZ_TARR)
#define SZ_SARR   (4*BB*3*NS)              /* 6,291,456 */
#define OFF_VALS  (OFF_SARR + SZ_SARR)     /* 7,864,320 */
#define OFF_ORDER (OFF_VALS + BB*BAND)
#define OFF_TXT   (OFF_ORDER + BB*BAND)    /* 7,864,720 */
#define SZ_TXT    (BB*BAND*NT)             /* 3,276,800 */
#define OFF_TXS   (OFF_TXT + SZ_TXT)
#define SZ_TXS    (BB*BAND*NS)             /* 13,107,200 */
#define OFF_WXT   (OFF_TXS + SZ_TXS)
#define OFF_WXS   (OFF_WXT + SZ_TXT)

/* ================= 1) per-(b,band) spatial mean ================= */
__global__ __launch_bounds__(256)
void meanpool_kernel(const float* __restrict__ t, const float* __restrict__ s,
                     float* __restrict__ tmean, float* __restrict__ smean) {
    __shared__ float red[256];
    int bi = blockIdx.x, tid = threadIdx.x;
    const float4* src; int n4; float* dst; float inv;
    if (bi < BB*BAND) {
        src = (const float4*)(t + (size_t)bi * NT);
        n4 = NT/4; dst = tmean + bi; inv = 1.0f / NT;
    } else {
        int p = bi - BB*BAND;
        src = (const float4*)(s + (size_t)p * NS);
        n4 = NS/4; dst = smean + p; inv = 1.0f / NS;
    }
    float acc = 0.f;
    for (int i = tid; i < n4; i += 256) { float4 v = src[i]; acc += (v.x + v.y) + (v.z + v.w); }
    red[tid] = acc; __syncthreads();
    for (int o = 128; o > 0; o >>= 1) { if (tid < o) red[tid] += red[tid + o]; __syncthreads(); }
    if (tid == 0) *dst = red[0] * inv;
}

/* ===== 2) fused small chain: proj -> attention -> w0 -> cm -> sort ===== */
__global__ __launch_bounds__(256)
void small_chain_kernel(const float* __restrict__ tmean, const float* __restrict__ smean,
                        const float* __restrict__ ipw, const float* __restrict__ ipb,
                        const float* __restrict__ ow,  const float* __restrict__ ob,
                        const float* __restrict__ cw,  const float* __restrict__ cb,
                        float* __restrict__ cm_ws, int* __restrict__ order_ws,
                        float* __restrict__ out_vals, float* __restrict__ out_order) {
    __shared__ float xt[BAND*EMB], xs[BAND*EMB];
    __shared__ float q[BAND*EMB], kk[BAND*EMB], vv[BAND*EMB];
    __shared__ float oo[BAND*EMB], w0[BAND*EMB];
    __shared__ float cml[BB*BAND*BAND];      /* 5000 floats */
    __shared__ float red[256];
    __shared__ float ysh[BB*BAND];

    int tid = threadIdx.x;
    float tmax = 0.0f;                       /* cm = relu·relu >= 0 */

    for (int b = 0; b < BB; ++b) {
        /* pooled projection -> xt/xs (BAND x EMB), xt[band][e] = pt[e*25+band] */
        for (int j = tid; j < EMB*BAND; j += 256) {
            int e = j / BAND, band = j % BAND;
            float at = cb[j], as = cb[j];
            for (int k = 0; k < BAND; ++k) {
                float w = cw[j*BAND + k];
                at += tmean[b*BAND + k] * w;
                as += smean[b*BAND + k] * w;
            }
            xt[band*EMB + e] = at; xs[band*EMB + e] = as;
        }
        __syncthreads();
        /* q from xs, k/v from xt */
        for (int j = tid; j < BAND*EMB; j += 256) {
            int i = j / EMB, e = j % EMB;
            float aq = ipb[e], ak = ipb[EMB + e], av = ipb[2*EMB + e];
            for (int c = 0; c < EMB; ++c) {
                float xsv = xs[i*EMB + c], xtv = xt[i*EMB + c];
                aq += xsv * ipw[e*EMB + c];
                ak += xtv * ipw[(EMB + e)*EMB + c];
                av += xtv * ipw[(2*EMB + e)*EMB + c];
            }
            q[j] = aq; kk[j] = ak; vv[j] = av;
        }
        __syncthreads();
        /* attention per (head, query) */
        if (tid < HEADS*BAND) {
            int h = tid / BAND, qi = tid % BAND;
            float sc[BAND]; float mx = -1e30f;
            for (int kj = 0; kj < BAND; ++kj) {
                float d = 0.f;
                for (int dd = 0; dd < HDIM; ++dd)
                    d += q[qi*EMB + h*HDIM + dd] * kk[kj*EMB + h*HDIM + dd];
                sc[kj] = d * 0.5f;                 /* 1/sqrt(4) */
                mx = fmaxf(mx, sc[kj]);
            }
            float se = 0.f;
            for (int kj = 0; kj < BAND; ++kj) { sc[kj] = __expf(sc[kj] - mx); se += sc[kj]; }
            float inv = 1.0f / se;
            for (int dd = 0; dd < HDIM; ++dd) {
                float a = 0.f;
                for (int kj = 0; kj < BAND; ++kj) a += sc[kj] * vv[kj*EMB + h*HDIM + dd];
                oo[qi*EMB + h*HDIM + dd] = a * inv;
            }
        }
        __syncthreads();
        /* w0 = relu(o @ ow.T + ob) */
        for (int j = tid; j < BAND*EMB; j += 256) {
            int i = j / EMB, e = j % EMB;
            float a = ob[e];
            for (int c = 0; c < EMB; ++c) a += oo[i*EMB + c] * ow[e*EMB + c];
            w0[j] = fmaxf(a, 0.f);
        }
        __syncthreads();
        /* cm_raw = w0 @ w0.T, track running max */
        for (int j = tid; j < BAND*BAND; j += 256) {
            int i = j / BAND, jj = j % BAND;
            float a = 0.f;
            for (int c = 0; c < EMB; ++c) a += w0[i*EMB + c] * w0[jj*EMB + c];
            cml[b*BAND*BAND + j] = a;
            tmax = fmaxf(tmax, a);
        }
        __syncthreads();
    }
    /* global max across all batches */
    red[tid] = tmax; __syncthreads();
    for (int o = 128; o > 0; o >>= 1) { if (tid < o) red[tid] = fmaxf(red[tid], red[tid + o]); __syncthreads(); }
    float ginv = 1.0f / red[0];
    /* normalize, zero diagonal, publish cm */
    for (int j = tid; j < BB*BAND*BAND; j += 256) {
        int l = j % (BAND*BAND); int i = l / BAND, jj = l % BAND;
        float v = (i == jj) ? 0.f : cml[j] * ginv;
        cml[j] = v; cm_ws[j] = v;
    }
    __syncthreads();
    /* y = row mean */
    if (tid < BB*BAND) {
        int b = tid / BAND, i = tid % BAND;
        float a = 0.f;
        for (int jj = 0; jj < BAND; ++jj) a += cml[b*BAND*BAND + i*BAND + jj];
        ysh[tid] = a * (1.0f / BAND);
    }
    __syncthreads();
    /* per-batch descending selection sort (swap-based, not stable on ties) */
    if (tid < BB) {
        int b = tid;
        float yl[BAND]; int idx[BAND];
        for (int i = 0; i < BAND; ++i) { yl[i] = ysh[b*BAND + i]; idx[i] = i; }
        for (int i = 0; i < BAND; ++i) {
            int best = i;
            for (int j = i + 1; j < BAND; ++j) if (yl[j] > yl[best]) best = j;
            float tv = yl[i]; yl[i] = yl[best]; yl[best] = tv;
            int ti = idx[i]; idx[i] = idx[best]; idx[best] = ti;
        }
        for (int i = 0; i < BAND; ++i) {
            out_vals[b*BAND + i]  = yl[i];
            out_order[b*BAND + i] = (float)idx[i];
            order_ws[b*BAND + i]  = idx[i];
        }
    }
}

/* ================= 3) verbatim tx copies (float4) ================= */
__global__ __launch_bounds__(256)
void copy_f4_kernel(const float4* __restrict__ src, float4* __restrict__ dst, int n4) {
    int i = blockIdx.x * 256 + threadIdx.x;
    if (i < n4) dst[i] = src[i];
}

/* ================= 4) gather ordered planes ================= */
__global__ __launch_bounds__(256)
void gather_planes_kernel(const float* __restrict__ t, const float* __restrict__ s,
                          const int* __restrict__ order,
                          float* __restrict__ out_t, float* __restrict__ out_s) {
    int bi = blockIdx.x;
    const float4* src; float4* dst; int n4;
    if (bi < 4*BB*3) {
        int g = bi / (BB*3), rem = bi % (BB*3), b = rem / 3, r = rem % 3;
        int band = order[b*BAND + g*3 + r];
        src = (const float4*)(t + (size_t)(b*BAND + band) * NT);
        dst = (float4*)(out_t + (size_t)(g*BB*3 + b*3 + r) * NT);
        n4 = NT/4;
    } else {
        int bj = bi - 4*BB*3;
        int g = bj / (BB*3), rem = bj % (BB*3), b = rem / 3, r = rem % 3;
        int band = order[b*BAND + g*3 + r];
        src = (const float4*)(s + (size_t)(b*BAND + band) * NS);
        dst = (float4*)(out_s + (size_t)(g*BB*3 + b*3 + r) * NS);
        n4 = NS/4;
    }
    for (int i = threadIdx.x; i < n4; i += 256) dst[i] = src[i];
}

/* ====== 5) wx = cm @ tx via V_WMMA_F32_16X16X4_F32 ======
 * Templated on NCOLS so every row stride is a compile-time constant:
 * all B loads / C stores become immediate-offset global ops off one base
 * pointer (max offset 23*NCOLS*4 < 2^23). Zero-padded A staged in LDS ->
 * branch-free ds_load_b64 fragments; B rows K>=25 folded to row 24 (A==0
 * there). Store guard restructured: wave-uniform mt branch + single
 * hf-predicated region instead of 8 per-row saveexec blocks.            */
template<int NCOLS>
__global__ __launch_bounds__(256)
void wx_wmma_kernel(const float* __restrict__ cm, const float* __restrict__ src,
                    float* __restrict__ dst) {
    __shared__ float cmp[32*28];
    constexpr int ntn = NCOLS >> 4;          /* 16-col tiles per row     */
    constexpr int bpb = (2*ntn) / (8*TPW);   /* blocks per batch (exact) */
    int tid = threadIdx.x;
    int b   = blockIdx.x / bpb;

    for (int j = tid; j < 32*28; j += 256) {
        int r = j / 28, c = j % 28;
        cmp[j] = (r < BAND && c < BAND) ? cm[b*BAND*BAND + r*BAND + c] : 0.f;
    }
    __syncthreads();

    int w    = tid >> 5;
    int lane = tid & 31;
    int hf   = lane >> 4;            /* half-wave selects K pair        */
    int l16  = lane & 15;
    int lbase = (blockIdx.x % bpb) * (8*TPW) + w * TPW;  /* tile group   */
    int mt   = lbase / ntn;          /* uniform across the TPW group    */
    int ntb  = lbase % ntn;
    int m    = mt*16 + l16;          /* A row (padded, always valid)    */

    v2f a[7];
#pragma unroll
    for (int i = 0; i < 7; ++i)      /* 8B-aligned LDS reads, no guards */
        a[i] = *(const v2f*)&cmp[m*28 + i*4 + hf*2];

    const float* srcb = src + (size_t)b * BAND * NCOLS;
    float*       dstb = dst + (size_t)b * BAND * NCOLS;

    for (int tt = 0; tt < TPW; ++tt) {
        int col = (ntb + tt)*16 + l16;
        /* one base per lane; all row offsets are compile-time immediates */
        const float* pb  = srcb + (size_t)(hf*2) * NCOLS + col;
        float        b24 = srcb[(size_t)24 * NCOLS + col];   /* shared K>=24 row */
        v2f bf[7];
#pragma unroll
        for (int i = 0; i < 6; ++i) {        /* rows i*4+hf*2 (+1), all <= 23 */
            bf[i].x = pb[(size_t)(i*4)     * NCOLS];
            bf[i].y = pb[(size_t)(i*4 + 1) * NCOLS];
        }
        bf[6].x = b24; bf[6].y = b24;        /* K=24..27: A==0 past 24 */

        v8f acc = {};
#pragma unroll
        for (int i = 0; i < 7; ++i)
            acc = __builtin_amdgcn_wmma_f32_16x16x4_f32(
                      false, a[i], false, bf[i], (short)0, acc, false, false);

        /* C/D layout: vgpr r, half -> M = mt*16 + hf*8 + r */
        float* pd = dstb + (size_t)(mt*16 + hf*8) * NCOLS + col;
        if (mt == 0) {                       /* rows 0..15: all valid   */
#pragma unroll
            for (int r = 0; r < 8; ++r) pd[(size_t)r * NCOLS] = acc[r];
        } else {
            if (hf == 0) {                   /* rows 16..23: all valid  */
#pragma unroll
                for (int r = 0; r < 8; ++r) pd[(size_t)r * NCOLS] = acc[r];
            } else {
                pd[0] = acc[0];              /* row 24 only             */
            }
        }
    }
}

extern "C" void kernel_launch(void* const* d_in, const int* in_sizes, int n_in,
                              void* d_out, int out_size, void* d_ws, size_t ws_size,
                              hipStream_t stream) {
    const float* t   = (const float*)d_in[0];
    const float* s   = (const float*)d_in[1];
    const float* ipw = (const float*)d_in[2];
    const float* ipb = (const float*)d_in[3];
    const float* ow  = (const float*)d_in[4];
    const float* ob  = (const float*)d_in[5];
    const float* cw  = (const float*)d_in[6];
    const float* cb  = (const float*)d_in[7];
    float* out = (float*)d_out;

    float* wsf   = (float*)d_ws;
    float* tmean = wsf;                 /* 200 */
    float* smean = wsf + 200;           /* 200 */
    float* cm_ws = wsf + 400;           /* 5000 */
    int*   ord   = (int*)(wsf + 5400);  /* 200 */

    meanpool_kernel<<<2*BB*BAND, 256, 0, stream>>>(t, s, tmean, smean);
    small_chain_kernel<<<1, 256, 0, stream>>>(tmean, smean, ipw, ipb, ow, ob, cw, cb,
                                              cm_ws, ord,
                                              out + OFF_VALS, out + OFF_ORDER);
    copy_f4_kernel<<<(SZ_TXT/4 + 255)/256, 256, 0, stream>>>(
        (const float4*)t, (float4*)(out + OFF_TXT), SZ_TXT/4);
    copy_f4_kernel<<<(SZ_TXS/4 + 255)/256, 256, 0, stream>>>(
        (const float4*)s, (float4*)(out + OFF_TXS), SZ_TXS/4);
    gather_planes_kernel<<<2*4*BB*3, 256, 0, stream>>>(t, s, ord,
                                                       out + OFF_TARR, out + OFF_SARR);
    /* wx_t: 8 batches * (2*1024 tiles)/(8 waves * 4 tiles) = 8*64 = 512 blocks */
    wx_wmma_kernel<NT><<<BB * ((2*(NT/16))/(8*TPW)), 256, 0, stream>>>(cm_ws, t, out + OFF_WXT);
    /* wx_s: 8 * (2*4096)/(32) = 8*256 = 2048 blocks */
    wx_wmma_kernel<NS><<<BB * ((2*(NS/16))/(8*TPW)), 256, 0, stream>>>(cm_ws, s, out + OFF_WXS);
}